// InferenceLayer_532575945010
// MI455X (gfx1250) — compile-verified
//
#include <hip/hip_runtime.h>
#include <hip/hip_bf16.h>

#define BB 64
#define TT 512
#define DD 1024
#define CC 32
#define BT (BB * TT)          // 32768 rows of the GEMM

typedef __attribute__((ext_vector_type(2))) float v2f;
typedef __attribute__((ext_vector_type(4))) float v4f;
typedef __attribute__((ext_vector_type(8))) float v8f;

// ---------------------------------------------------------------------------
// Kernel 1: logits = vectors @ W^T + b via V_WMMA_F32_16X16X4_F32
// grid = 256 blocks x 256 threads (8 waves). Wave w of block handles
// M-tile rows [(blk*8+w)*16, +16) and both N-tiles (classes 0-15, 16-31).
// K chunked by 64: W chunk staged block-wide in LDS, A chunk per-wave in LDS,
// rows padded to 68 floats to avoid 16-way LDS bank conflicts.
// ---------------------------------------------------------------------------
__global__ __launch_bounds__(256) void logits_wmma_kernel(
    const float* __restrict__ vectors, const float* __restrict__ W,
    const float* __restrict__ bias, float* __restrict__ logits) {
  __shared__ float ldsW[CC * 68];          // 32 x 64 chunk (padded)
  __shared__ float ldsA[8 * 16 * 68];      // per-wave 16 x 64 chunk (padded)

  const int tid  = threadIdx.x;
  const int lane = tid & 31;
  const int wave = tid >> 5;
  const int half = lane >> 4;              // 0: lanes 0-15, 1: lanes 16-31
  const int l16  = lane & 15;
  const int mbase = (blockIdx.x * 8 + wave) * 16;

  float* LA = &ldsA[wave * 16 * 68];
  v8f acc0 = {};
  v8f acc1 = {};

  for (int kc = 0; kc < DD; kc += 64) {
    // prefetch next vectors chunk while we compute (global_prefetch_b8)
    if (kc + 64 < DD)
      __builtin_prefetch(&vectors[(size_t)(mbase + l16) * DD + kc + 64], 0, 1);

    __syncthreads();   // previous chunk fully consumed
    // stage W chunk: 32 rows x 64 floats = 512 float4, 256 threads x 2
    #pragma unroll
    for (int i = 0; i < 2; ++i) {
      int idx = i * 256 + tid;             // 0..511
      int row = idx >> 4;                  // class 0..31
      int c4  = idx & 15;                  // float4 column
      v4f w = *(const v4f*)(&W[(size_t)row * DD + kc + c4 * 4]);
      *(v4f*)(&ldsW[row * 68 + c4 * 4]) = w;
    }
    // stage per-wave A chunk: 16 rows x 64 floats = 256 float4, 32 lanes x 8
    #pragma unroll
    for (int i = 0; i < 8; ++i) {
      int idx = i * 32 + lane;             // 0..255
      int row = idx >> 4;                  // 0..15
      int c4  = idx & 15;
      v4f a = *(const v4f*)(&vectors[(size_t)(mbase + row) * DD + kc + c4 * 4]);
      *(v4f*)(&LA[row * 68 + c4 * 4]) = a;
    }
    __syncthreads();   // ldsW visible to all waves

    // 16 x (K=4) WMMA steps per chunk, two N-tiles each
    #pragma unroll
    for (int kk = 0; kk < 64; kk += 4) {
      // A 16x4 layout: VGPR v holds K = v + 2*half, row M = l16
      v2f a  = *(const v2f*)(&LA[l16 * 68 + kk + 2 * half]);
      // B 4x16 layout: VGPR v holds K = v + 2*half, col N = l16; B[k][n] = W[n][k]
      v2f b0 = *(const v2f*)(&ldsW[l16 * 68 + kk + 2 * half]);
      v2f b1 = *(const v2f*)(&ldsW[(16 + l16) * 68 + kk + 2 * half]);
      acc0 = __builtin_amdgcn_wmma_f32_16x16x4_f32(false, a, false, b0,
                                                   (short)0, acc0, false, false);
      acc1 = __builtin_amdgcn_wmma_f32_16x16x4_f32(false, a, false, b1,
                                                   (short)0, acc1, false, false);
    }
  }

  // C/D layout: VGPR v holds row M = v + 8*half, col N = l16
  #pragma unroll
  for (int v = 0; v < 8; ++v) {
    int m = mbase + v + 8 * half;
    logits[(size_t)m * CC + l16]      = acc0[v] + bias[l16];
    logits[(size_t)m * CC + 16 + l16] = acc1[v] + bias[16 + l16];
  }
}

// ---------------------------------------------------------------------------
// Kernel 2: CRF forward (log-Z), Viterbi (+backpointers), gold score pieces,
// backtrace. One wave32 per batch; lane j = class j (C == wave32 width).
// ---------------------------------------------------------------------------
__global__ __launch_bounds__(32) void crf_kernel(
    const float* __restrict__ logits, const int* __restrict__ mask,
    const int* __restrict__ targets, const float* __restrict__ transitions,
    const float* __restrict__ startt, const float* __restrict__ endt,
    float* __restrict__ tags_out, float* __restrict__ path_scores,
    float* __restrict__ partials, unsigned char* __restrict__ bp_ws) {
  const int b = blockIdx.x;
  const int j = threadIdx.x;                       // class index
  const float* lg = logits + (size_t)b * TT * CC;
  const int* mk = mask + b * TT;
  const int* tg = targets + b * TT;
  unsigned char* bp = bp_ws + (size_t)b * (TT - 1) * CC;

  // preload transition column j into registers (trans[i][j])
  float trc[CC];
  #pragma unroll
  for (int i = 0; i < CC; ++i) trc[i] = transitions[i * CC + j];

  __shared__ float sA[CC], sD[CC];
  float alpha = startt[j] + lg[j];
  float delta = alpha;

  for (int t = 1; t < TT; ++t) {
    sA[j] = alpha;
    sD[j] = delta;
    __syncthreads();
    float lt = lg[t * CC + j];
    int mt = mk[t];
    // forward: logsumexp_i(alpha[i] + trans[i][j]) + logit
    float m = -3.4e38f;
    #pragma unroll
    for (int i = 0; i < CC; ++i) m = fmaxf(m, sA[i] + trc[i]);
    float s = 0.f;
    #pragma unroll
    for (int i = 0; i < CC; ++i) s += __expf(sA[i] + trc[i] - m);
    float na = m + __logf(s) + lt;
    // viterbi: max/argmax_i(delta[i] + trans[i][j]) (first-max tiebreak)
    float best = -3.4e38f;
    int bi = 0;
    #pragma unroll
    for (int i = 0; i < CC; ++i) {
      float v = sD[i] + trc[i];
      if (v > best) { best = v; bi = i; }
    }
    float nd = best + lt;
    alpha = mt ? na : alpha;
    delta = mt ? nd : delta;
    bp[(t - 1) * CC + j] = (unsigned char)(mt ? bi : j);
    __syncthreads();
  }

  // log_Z = logsumexp_j(alpha + end) via wave32 shuffles
  float x = alpha + endt[j];
  float mm = x;
  for (int off = 16; off > 0; off >>= 1) mm = fmaxf(mm, __shfl_xor(mm, off, 32));
  float se = __expf(x - mm);
  for (int off = 16; off > 0; off >>= 1) se += __shfl_xor(se, off, 32);
  float logZ = mm + __logf(se);

  // viterbi final: max + first-index argmax over lanes
  float fv = delta + endt[j];
  int fi = j;
  for (int off = 16; off > 0; off >>= 1) {
    float ov = __shfl_xor(fv, off, 32);
    int oi = __shfl_xor(fi, off, 32);
    if (ov > fv || (ov == fv && oi < fi)) { fv = ov; fi = oi; }
  }

  // gold score pieces, lane-parallel over time
  float emit_s = 0.f, trans_s = 0.f;
  int msum = 0;
  for (int t = j; t < TT; t += CC) {
    int mt = mk[t];
    msum += mt;
    int tt = tg[t];
    if (t < TT - 1) emit_s += lg[t * CC + tt] * (float)mt;     // emit[:, :-1]*mask[:, :-1]
    if (t >= 1) trans_s += transitions[tg[t - 1] * CC + tt] * (float)mt;
  }
  for (int off = 16; off > 0; off >>= 1) {
    emit_s  += __shfl_xor(emit_s, off, 32);
    trans_s += __shfl_xor(trans_s, off, 32);
    msum    += __shfl_xor(msum, off, 32);
  }

  if (j == 0) {
    int last_idx = msum - 1;
    int last_tg = tg[last_idx];
    float score = startt[tg[0]] + trans_s + emit_s + endt[last_tg] +
                  lg[(TT - 1) * CC + last_tg] * (float)mk[TT - 1];
    partials[b * 3 + 0] = score;
    partials[b * 3 + 1] = logZ;
    partials[b * 3 + 2] = (float)msum;
    path_scores[b] = fv;
    // backtrace
    int tag = fi;
    tags_out[(size_t)b * TT + (TT - 1)] = (float)tag;
    for (int t = TT - 1; t >= 1; --t) {
      tag = bp[(t - 1) * CC + tag];
      tags_out[(size_t)b * TT + (t - 1)] = (float)tag;
    }
  }
}

// ---------------------------------------------------------------------------
// Kernel 3: loss = -sum_b(score - logZ) / sum(mask)
// ---------------------------------------------------------------------------
__global__ __launch_bounds__(64) void loss_kernel(const float* __restrict__ partials,
                                                  float* __restrict__ out) {
  __shared__ float ssl[BB], sm[BB];
  int b = threadIdx.x;
  ssl[b] = partials[b * 3 + 0] - partials[b * 3 + 1];
  sm[b]  = partials[b * 3 + 2];
  __syncthreads();
  for (int off = 32; off > 0; off >>= 1) {
    if (b < off) { ssl[b] += ssl[b + off]; sm[b] += sm[b + off]; }
    __syncthreads();
  }
  if (b == 0) out[0] = -ssl[0] / sm[0];
}

extern "C" void kernel_launch(void* const* d_in, const int* in_sizes, int n_in,
                              void* d_out, int out_size, void* d_ws, size_t ws_size,
                              hipStream_t stream) {
  const float* vectors     = (const float*)d_in[0];
  const int*   mask        = (const int*)d_in[1];
  const int*   targets     = (const int*)d_in[2];
  const float* W           = (const float*)d_in[3];
  const float* bias        = (const float*)d_in[4];
  const float* transitions = (const float*)d_in[5];
  const float* startt      = (const float*)d_in[6];
  const float* endt        = (const float*)d_in[7];

  float* out         = (float*)d_out;
  float* logits      = out + 1;                       // B*T*C
  float* tags_out    = logits + (size_t)BT * CC;      // B*T
  float* path_scores = tags_out + BT;                 // B

  float* partials = (float*)d_ws;                     // 64 x {score, logZ, msum}
  unsigned char* bp = (unsigned char*)d_ws + 1024;    // 64 * 511 * 32 bytes

  logits_wmma_kernel<<<BT / 128, 256, 0, stream>>>(vectors, W, bias, logits);
  crf_kernel<<<BB, 32, 0, stream>>>(logits, mask, targets, transitions, startt,
                                    endt, tags_out, path_scores, partials, bp);
  loss_kernel<<<1, BB, 0, stream>>>(partials, out);
}